// MoELayer_26182120637034
// MI455X (gfx1250) — compile-verified
//
#include <hip/hip_runtime.h>
#include <hip/hip_bf16.h>

// Problem constants (from reference: B=4,S=2048,D=1024,E=8,O=1024,K=2)
#define T_TOKENS 8192
#define DDIM     1024
#define NEXP     8
#define ODIM     1024
#define TOPK     2
#define MTILE    32     // tokens per block (2 x 16-row WMMA tiles)
#define MAXTILES 528    // sum_e ceil(cnt_e/32) <= 16384/32 + 8 = 520

typedef float v2f __attribute__((ext_vector_type(2)));
typedef float v8f __attribute__((ext_vector_type(8)));

// ---------------------------------------------------------------- gating ----
__global__ void __launch_bounds__(256)
moe_gate(const float* __restrict__ x, const float* __restrict__ Wg,
         int* __restrict__ sel, float* __restrict__ wgt, int* __restrict__ cnt)
{
    __shared__ float wg[NEXP * DDIM];            // 32 KB of 320 KB WGP LDS
    const int tid = threadIdx.x;
    for (int i = tid; i < NEXP * DDIM / 4; i += 256)
        reinterpret_cast<float4*>(wg)[i] = reinterpret_cast<const float4*>(Wg)[i];
    __syncthreads();

    const int t = blockIdx.x * 256 + tid;
    if (t >= T_TOKENS) return;
    const float* __restrict__ xr = x + (size_t)t * DDIM;

    float acc[NEXP];
#pragma unroll
    for (int e = 0; e < NEXP; ++e) acc[e] = 0.f;

    for (int d = 0; d < DDIM; d += 4) {
        const float4 xv = *reinterpret_cast<const float4*>(xr + d);
#pragma unroll
        for (int e = 0; e < NEXP; ++e) {
            const float4 wv = *reinterpret_cast<const float4*>(&wg[e * DDIM + d]);
            acc[e] += xv.x * wv.x + xv.y * wv.y + xv.z * wv.z + xv.w * wv.w;
        }
    }

    // top-2 (strict > keeps lowest index on ties, matching jax.lax.top_k)
    float v0 = -3.4e38f; int e0 = 0;
#pragma unroll
    for (int e = 0; e < NEXP; ++e) { if (acc[e] > v0) { v0 = acc[e]; e0 = e; } }
    float v1 = -3.4e38f; int e1 = 0;
#pragma unroll
    for (int e = 0; e < NEXP; ++e) { if (e != e0 && acc[e] > v1) { v1 = acc[e]; e1 = e; } }

    const float z  = __expf(v1 - v0);            // softmax over the 2 selected
    const float w0 = 1.f / (1.f + z);
    const float w1 = z * w0;

    sel[2 * t]     = e0;  sel[2 * t + 1] = e1;
    wgt[2 * t]     = w0;  wgt[2 * t + 1] = w1;
    atomicAdd(&cnt[e0], 1);
    atomicAdd(&cnt[e1], 1);
}

// ------------------------------------------------------------------ plan ----
__global__ void moe_plan(const int* __restrict__ cnt, int* __restrict__ off,
                         int* __restrict__ cursor, int* __restrict__ tileE,
                         int* __restrict__ tileR, int* __restrict__ nTiles)
{
    if (threadIdx.x != 0 || blockIdx.x != 0) return;
    int o = 0, idx = 0;
    for (int e = 0; e < NEXP; ++e) {
        off[e] = o; cursor[e] = 0;
        const int rt = (cnt[e] + MTILE - 1) / MTILE;
        for (int i = 0; i < rt && idx < MAXTILES; ++i) { tileE[idx] = e; tileR[idx] = i; ++idx; }
        o += cnt[e];
    }
    *nTiles = idx;
}

// --------------------------------------------------------------- scatter ----
__global__ void __launch_bounds__(256)
moe_scatter(const int* __restrict__ sel, const float* __restrict__ wgt,
            const int* __restrict__ off, int* __restrict__ cursor,
            int* __restrict__ slotTok, float* __restrict__ slotW)
{
    const int t = blockIdx.x * 256 + threadIdx.x;
    if (t >= T_TOKENS) return;
#pragma unroll
    for (int k = 0; k < TOPK; ++k) {
        const int e = sel[2 * t + k];
        const int p = off[e] + atomicAdd(&cursor[e], 1);
        slotTok[p] = t;
        slotW[p]   = wgt[2 * t + k];
    }
}

// ------------------------------------------------- bias init of output ------
__global__ void __launch_bounds__(256)
moe_bias(const float* __restrict__ be, const int* __restrict__ sel,
         const float* __restrict__ wgt, float* __restrict__ out)
{
    const int g  = blockIdx.x * 256 + threadIdx.x;     // one float4 per thread
    const int t  = g / (ODIM / 4);
    const int o4 = (g % (ODIM / 4)) * 4;
    if (t >= T_TOKENS) return;
    const int   e0 = sel[2 * t], e1 = sel[2 * t + 1];
    const float w0 = wgt[2 * t], w1 = wgt[2 * t + 1];
    const float4 b0 = *reinterpret_cast<const float4*>(be + (size_t)e0 * ODIM + o4);
    const float4 b1 = *reinterpret_cast<const float4*>(be + (size_t)e1 * ODIM + o4);
    float4 r;
    r.x = w0 * b0.x + w1 * b1.x;  r.y = w0 * b0.y + w1 * b1.y;
    r.z = w0 * b0.z + w1 * b1.z;  r.w = w0 * b0.w + w1 * b1.w;
    *reinterpret_cast<float4*>(out + (size_t)t * ODIM + o4) = r;
}

// ------------------------------------------------ expert GEMM via WMMA ------
// Block: 128 threads (4 wave32). Tile: 32 tokens x 64 outputs.
// Each wave owns one 16-col slice with TWO 16x16 accumulators (rows 0-15 and
// 16-31), so every B fetch feeds two V_WMMA_F32_16X16X4_F32.
// Invalid (padding) rows are clamped to token 0: they produce finite garbage
// only in their own C row, whose store is skipped in the epilogue, so the
// staging loads are unconditional (no exec-mask churn in the hot loop).
#define CK   32     // K chunk staged in LDS
#define APAD 36     // row stride (floats): 144B rows -> 16B-aligned b128 LDS ops,
                    // m*36 mod 64 hits 16 distinct banks, K-pairs stay 8B aligned

__global__ void __launch_bounds__(128)
moe_gemm(const float* __restrict__ x, const float* __restrict__ We,
         const int* __restrict__ slotTok, const float* __restrict__ slotW,
         const int* __restrict__ tileE, const int* __restrict__ tileR,
         const int* __restrict__ nTiles, const int* __restrict__ cnt,
         const int* __restrict__ off, float* __restrict__ out)
{
    __shared__ float A[MTILE * APAD];            // 32 x 32 f32 chunk (padded)
    __shared__ int   stok[MTILE];
    __shared__ float sw[MTILE];

    const int tile = blockIdx.y;
    if (tile >= *nTiles) return;                 // uniform early-exit
    const int e    = tileE[tile];
    const int r0   = tileR[tile] * MTILE;
    const int n    = cnt[e];
    const int base = off[e];

    const int tid = threadIdx.x;
    if (tid < MTILE) {
        const int  idx = r0 + tid;
        const bool v   = idx < n;
        stok[tid] = v ? slotTok[base + idx] : -1;
        sw[tid]   = v ? slotW[base + idx]   : 0.f;
    }
    __syncthreads();

    const int lane = tid & 31;
    const int wv   = tid >> 5;                   // wave id 0..3
    const int nn   = lane & 15;                  // N col (B/C/D) == M row (A)
    const int ocol = blockIdx.x * 64 + wv * 16 + nn;
    const float* __restrict__ wr = We + ((size_t)e * ODIM + ocol) * DDIM;

    // A-tile loader: each thread fills rows r and r+16, cols [c, c+4)
    const int ldr = tid >> 3;                    // 0..15
    const int ldc = (tid & 7) << 2;              // 0,4,..,28
    const int t0  = stok[ldr];                   // hoisted: constant over K-loop
    const int t1  = stok[ldr + 16];
    const float* __restrict__ xr0 = x + (size_t)(t0 < 0 ? 0 : t0) * DDIM + ldc;
    const float* __restrict__ xr1 = x + (size_t)(t1 < 0 ? 0 : t1) * DDIM + ldc;
    float4* const ap0 = reinterpret_cast<float4*>(&A[ldr * APAD + ldc]);
    float4* const ap1 = reinterpret_cast<float4*>(&A[(ldr + 16) * APAD + ldc]);

    const int koff  = (lane >> 4) << 1;          // 0 or 2 (A/B K-halves)
    const int arow0 = nn * APAD + koff;
    const int arow1 = (nn + 16) * APAD + koff;

    v8f c0 = {};
    v8f c1 = {};
    for (int kk = 0; kk < DDIM; kk += CK) {
        // cooperative stage: 32 tokens x 32 k-values of gathered X (no predication)
        *ap0 = *reinterpret_cast<const float4*>(xr0 + kk);
        *ap1 = *reinterpret_cast<const float4*>(xr1 + kk);
        __syncthreads();

        // V_WMMA_F32_16X16X4_F32: lanes0-15 hold K0/K1, lanes16-31 hold K2/K3
#pragma unroll
        for (int j = 0; j < 8; ++j) {
            const v2f b = *reinterpret_cast<const v2f*>(wr + kk + j * 4 + koff);
            v2f a0, a1;
            a0.x = A[arow0 + j * 4];  a0.y = A[arow0 + j * 4 + 1];
            a1.x = A[arow1 + j * 4];  a1.y = A[arow1 + j * 4 + 1];
            c0 = __builtin_amdgcn_wmma_f32_16x16x4_f32(false, a0, false, b,
                                                       (short)0, c0, false, false);
            c1 = __builtin_amdgcn_wmma_f32_16x16x4_f32(false, a1, false, b,
                                                       (short)0, c1, false, false);
        }
        __syncthreads();
    }

    // D layout: VGPR r -> M = r (lanes0-15) / r+8 (lanes16-31); N = lane&15
    const int mhi = (lane < 16) ? 0 : 8;
#pragma unroll
    for (int r = 0; r < 8; ++r) {
        const int m  = r + mhi;
        const int tt = stok[m];
        if (tt >= 0)
            unsafeAtomicAdd(out + (size_t)tt * ODIM + ocol, sw[m] * c0[r]);
        const int m2  = m + 16;
        const int tt2 = stok[m2];
        if (tt2 >= 0)
            unsafeAtomicAdd(out + (size_t)tt2 * ODIM + ocol, sw[m2] * c1[r]);
    }
}

// ---------------------------------------------------------------- launch ----
extern "C" void kernel_launch(void* const* d_in, const int* in_sizes, int n_in,
                              void* d_out, int out_size, void* d_ws, size_t ws_size,
                              hipStream_t stream) {
    const float* x  = (const float*)d_in[0];
    const float* Wg = (const float*)d_in[1];
    const float* We = (const float*)d_in[2];
    const float* be = (const float*)d_in[3];
    float* out = (float*)d_out;
    (void)in_sizes; (void)n_in; (void)out_size; (void)ws_size;

    char* wsb = (char*)d_ws;
    size_t o = 0;
    auto carve = [&](size_t bytes) -> void* {
        void* p = wsb + o;
        o += (bytes + 255) & ~(size_t)255;
        return p;
    };
    int*   cnt     = (int*)  carve(NEXP * sizeof(int));
    int*   cursor  = (int*)  carve(NEXP * sizeof(int));
    int*   off     = (int*)  carve(NEXP * sizeof(int));
    int*   nTiles  = (int*)  carve(sizeof(int));
    int*   sel     = (int*)  carve((size_t)T_TOKENS * 2 * sizeof(int));
    float* wgt     = (float*)carve((size_t)T_TOKENS * 2 * sizeof(float));
    int*   slotTok = (int*)  carve((size_t)T_TOKENS * 2 * sizeof(int));
    float* slotW   = (float*)carve((size_t)T_TOKENS * 2 * sizeof(float));
    int*   tileE   = (int*)  carve(MAXTILES * sizeof(int));
    int*   tileR   = (int*)  carve(MAXTILES * sizeof(int));

    hipMemsetAsync(cnt, 0, NEXP * sizeof(int), stream);

    moe_gate   <<<T_TOKENS / 256, 256, 0, stream>>>(x, Wg, sel, wgt, cnt);
    moe_plan   <<<1, 32, 0, stream>>>(cnt, off, cursor, tileE, tileR, nTiles);
    moe_scatter<<<T_TOKENS / 256, 256, 0, stream>>>(sel, wgt, off, cursor, slotTok, slotW);
    moe_bias   <<<(T_TOKENS * (ODIM / 4)) / 256, 256, 0, stream>>>(be, sel, wgt, out);

    dim3 g(ODIM / 64, MAXTILES);
    moe_gemm<<<g, 128, 0, stream>>>(x, We, slotTok, slotW, tileE, tileR,
                                    nTiles, cnt, off, out);
}